// GNNRecommender_13657996001410
// MI455X (gfx1250) — compile-verified
//
#include <hip/hip_runtime.h>
#include <math.h>

typedef __attribute__((ext_vector_type(2))) float v2f;
typedef __attribute__((ext_vector_type(8))) float v8f;

__device__ __forceinline__ void atomic_add_f32(float* p, float v) {
  __hip_atomic_fetch_add(p, v, __ATOMIC_RELAXED, __HIP_MEMORY_SCOPE_AGENT);
}

// ---------------- degree accumulation: deg[col] += w ----------------
__global__ void deg_kernel(const int* __restrict__ cols,
                           const float* __restrict__ ew,
                           float* __restrict__ deg, int E) {
  int e = blockIdx.x * blockDim.x + threadIdx.x;
  if (e >= E) return;
  atomic_add_f32(&deg[cols[e]], ew[e]);
}

// deg -> deg^{-1/2} in place (self-loop weight 1.0 folded in here)
__global__ void dinv_kernel(float* __restrict__ deg, int N) {
  int i = blockIdx.x * blockDim.x + threadIdx.x;
  if (i >= N) return;
  float d = deg[i] + 1.0f;
  deg[i] = (d > 0.0f) ? rsqrtf(d) : 0.0f;
}

// ---------------- fp32 WMMA GEMM: C[N x NCOLS] = A[N x 64] * B[64 x NCOLS] --
// NCOLS is compile-time so every load uses an immediate offset.
// one 16x16 output tile per wave; blockDim.x = 32 * (NCOLS/16); grid = N/16
template <int NCOLS>
__global__ void gemm_wmma_f32(const float* __restrict__ A,
                              const float* __restrict__ B,
                              float* __restrict__ C) {
  const int m0   = blockIdx.x * 16;
  const int wid  = threadIdx.x >> 5;
  const int lane = threadIdx.x & 31;
  const int n0   = wid * 16;
  const int ln   = lane & 15;   // m index for A, n index for B/C
  const int half = lane >> 4;   // K-pair / row-half selector

  v8f acc = {};
  const float* aptr = A + (size_t)(m0 + ln) * 64 + 2 * half;      // a[k], a[k+1]
  const float* bptr = B + (size_t)(2 * half) * NCOLS + n0 + ln;   // b[k*NCOLS], +NCOLS

#pragma unroll
  for (int k = 0; k < 64; k += 4) {
    v2f a, b;
    a.x = aptr[k];                 // A[m][k + 2*half]
    a.y = aptr[k + 1];             // A[m][k + 2*half + 1]
    b.x = bptr[k * NCOLS];         // B[k + 2*half][n]
    b.y = bptr[k * NCOLS + NCOLS]; // B[k + 2*half + 1][n]
    acc = __builtin_amdgcn_wmma_f32_16x16x4_f32(
        /*neg_a=*/false, a, /*neg_b=*/false, b,
        /*c_mod=*/(short)0, acc, /*reuse_a=*/false, /*reuse_b=*/false);
  }

  float* cptr = C + (size_t)(m0 + 8 * half) * NCOLS + n0 + ln;
#pragma unroll
  for (int r = 0; r < 8; ++r) {
    cptr[(size_t)r * NCOLS] = acc[r];   // row m0 + r + 8*half
  }
}

// ---------------- edge aggregation: out[col] += x[row] * norm --------------
// norm = dinv[row] * w * dinv[col]; virtual edges [E, E+N) are self-loops w=1
template <int CH>
__global__ void aggregate_kernel(const float* __restrict__ x,
                                 const int* __restrict__ rows,
                                 const int* __restrict__ cols,
                                 const float* __restrict__ ew,
                                 const float* __restrict__ dinv,
                                 float* __restrict__ out,
                                 int E, int N) {
  const int VPE = CH / 4;  // float4 chunks per edge
  long long tid = (long long)blockIdx.x * blockDim.x + threadIdx.x;
  long long total = (long long)(E + N) * VPE;
  if (tid >= total) return;
  long long e = tid / VPE;
  int c4 = (int)(tid % VPE);

  int r, c; float w;
  if (e < E) { r = rows[e]; c = cols[e]; w = ew[e]; }
  else       { r = c = (int)(e - E);     w = 1.0f; }

  float nrm = dinv[r] * w * dinv[c];
  const float4 xv = *(const float4*)(x + (long long)r * CH + c4 * 4);
  float* dst = out + (long long)c * CH + c4 * 4;
  atomic_add_f32(dst + 0, xv.x * nrm);
  atomic_add_f32(dst + 1, xv.y * nrm);
  atomic_add_f32(dst + 2, xv.z * nrm);
  atomic_add_f32(dst + 3, xv.w * nrm);
}

// ---------------- bias + relu, in place over [N x 64] (float4) -------------
__global__ void bias_relu_kernel(float* __restrict__ x,
                                 const float* __restrict__ b, int N) {
  long long tid = (long long)blockIdx.x * blockDim.x + threadIdx.x;
  if (tid >= (long long)N * 16) return;
  int c4 = (int)(tid & 15);
  float4 v  = ((float4*)x)[tid];
  float4 bb = ((const float4*)b)[c4];
  v.x = fmaxf(v.x + bb.x, 0.0f);
  v.y = fmaxf(v.y + bb.y, 0.0f);
  v.z = fmaxf(v.z + bb.z, 0.0f);
  v.w = fmaxf(v.w + bb.w, 0.0f);
  ((float4*)x)[tid] = v;
}

// ---------------- +b2 then row-wise L2 normalize over 32 channels ----------
// one wave32 per row; channel = lane
__global__ void l2norm_kernel(const float* __restrict__ x,
                              const float* __restrict__ b2,
                              float* __restrict__ out, int N) {
  int row  = blockIdx.x * (blockDim.x >> 5) + (threadIdx.x >> 5);
  int lane = threadIdx.x & 31;
  if (row >= N) return;
  float v = x[(long long)row * 32 + lane] + b2[lane];
  float ss = v * v;
#pragma unroll
  for (int off = 16; off > 0; off >>= 1)
    ss += __shfl_xor(ss, off, 32);
  float l2 = sqrtf(ss);
  float scale = 1.0f / fmaxf(l2, 1e-12f);
  out[(long long)row * 32 + lane] = v * scale;
}

extern "C" void kernel_launch(void* const* d_in, const int* in_sizes, int n_in,
                              void* d_out, int out_size, void* d_ws, size_t ws_size,
                              hipStream_t stream) {
  const int*   edge_index = (const int*)d_in[0];   // [2, E] flat
  const float* ew         = (const float*)d_in[1]; // [E]
  const float* emb        = (const float*)d_in[2]; // [N, 64]
  const float* W1         = (const float*)d_in[3]; // [64, 64]
  const float* b1         = (const float*)d_in[4]; // [64]
  const float* W2         = (const float*)d_in[5]; // [64, 32]
  const float* b2         = (const float*)d_in[6]; // [32]

  const int E = in_sizes[0] / 2;
  const int N = in_sizes[2] / 64;
  const int* rows = edge_index;
  const int* cols = edge_index + E;

  // workspace layout (floats): dinv[N] | xw1[N*64] | out1[N*64] | xw2[N*32] | out2[N*32]
  float* ws   = (float*)d_ws;
  float* dinv = ws;
  float* xw1  = dinv + (size_t)N;
  float* out1 = xw1  + (size_t)N * 64;
  float* xw2  = out1 + (size_t)N * 64;
  float* out2 = xw2  + (size_t)N * 32;

  hipMemsetAsync(dinv, 0, sizeof(float) * (size_t)N, stream);
  hipMemsetAsync(out1, 0, sizeof(float) * (size_t)N * 64, stream);
  hipMemsetAsync(out2, 0, sizeof(float) * (size_t)N * 32, stream);

  // symmetric normalization factors
  deg_kernel<<<(E + 255) / 256, 256, 0, stream>>>(cols, ew, dinv, E);
  dinv_kernel<<<(N + 255) / 256, 256, 0, stream>>>(dinv, N);

  // layer 1: GEMM (WMMA) -> edge aggregate -> bias+relu
  gemm_wmma_f32<64><<<N / 16, 128, 0, stream>>>(emb, W1, xw1);
  {
    long long total = (long long)(E + N) * 16;
    aggregate_kernel<64><<<(unsigned)((total + 255) / 256), 256, 0, stream>>>(
        xw1, rows, cols, ew, dinv, out1, E, N);
  }
  bias_relu_kernel<<<(unsigned)(((long long)N * 16 + 255) / 256), 256, 0, stream>>>(out1, b1, N);

  // layer 2: GEMM (WMMA) -> edge aggregate
  gemm_wmma_f32<32><<<N / 16, 64, 0, stream>>>(out1, W2, xw2);
  {
    long long total = (long long)(E + N) * 8;
    aggregate_kernel<32><<<(unsigned)((total + 255) / 256), 256, 0, stream>>>(
        xw2, rows, cols, ew, dinv, out2, E, N);
  }

  // +b2 and row-wise L2 normalize into d_out
  l2norm_kernel<<<(N + 7) / 8, 256, 0, stream>>>(out2, b2, (float*)d_out, N);
}